// SlicedTransport_34926674051675
// MI455X (gfx1250) — compile-verified
//
#include <hip/hip_runtime.h>

typedef float v2f __attribute__((ext_vector_type(2)));
typedef float v8f __attribute__((ext_vector_type(8)));
typedef int   v4i __attribute__((vector_size(16)));   // matches async builtin's param type

#define GLOBAL_AS __attribute__((address_space(1)))
#define LDS_AS    __attribute__((address_space(3)))

#if __has_builtin(__builtin_amdgcn_global_load_async_to_lds_b128)
#define HAVE_ASYNC_LDS 1
#else
#define HAVE_ASYNC_LDS 0
#warning "gfx1250 async-to-LDS builtin unavailable; using sync register staging"
#endif

namespace {

constexpr int NCOL = 512;   // columns of every matrix, also inner K of both GEMMs
constexpr int MK   = 200;   // spline knots per dimension

__device__ __forceinline__ void wait_async0() {
#if __has_builtin(__builtin_amdgcn_s_wait_asynccnt)
    __builtin_amdgcn_s_wait_asynccnt(0);
#else
    asm volatile("s_wait_asynccnt 0x0" ::: "memory");
#endif
}

#if HAVE_ASYNC_LDS
__device__ __forceinline__ void async_cp16(const float* g, float* l) {
    __builtin_amdgcn_global_load_async_to_lds_b128(
        (GLOBAL_AS v4i*)const_cast<float*>(g),
        (LDS_AS v4i*)l, 0, 0);
}
#endif

// 512x512 transpose: At[n][k] = A[k][n]  (1 MB, negligible runtime)
__global__ __launch_bounds__(256)
void transpose512(const float* __restrict__ src, float* __restrict__ dst)
{
    __shared__ float t[32][33];
    const int bx = blockIdx.x * 32, by = blockIdx.y * 32;
    const int x = threadIdx.x, y0 = threadIdx.y;   // block (32, 8)
#pragma unroll
    for (int j = 0; j < 32; j += 8)
        t[y0 + j][x] = src[(size_t)(by + y0 + j) * NCOL + bx + x];
    __syncthreads();
#pragma unroll
    for (int j = 0; j < 32; j += 8)
        dst[(size_t)(bx + y0 + j) * NCOL + by + x] = t[x][y0 + j];
}

// C(N x 512) = X @ B.
// TRANSB=1 : B[k][n] = W[n*512 + k]  (W holds B^T; contiguous along k -> async + b64 frags)
// TRANSB=0 : B[k][n] = W[k*512 + n]  (k-major fallback)
// EPI==0 : C = acc      EPI==1 : C = D - acc
template<bool TRANSB, int EPI>
__global__ __launch_bounds__(256)
void gemm512_wmma(const float* __restrict__ X,
                  const float* __restrict__ W,
                  const float* __restrict__ D,
                  float* __restrict__ C)
{
    constexpr int BM = 128, BN = 128, BK = 16;
    constexpr int XS  = BK + 4;    // X tile stride (floats): 16B-aligned rows, conflict-free frags
    constexpr int BS  = BN + 8;    // B tile stride, k-major layout (TRANSB=0)
    constexpr int BTS = BK + 4;    // B tile stride, n-major layout (TRANSB=1)
    constexpr int BSZ = TRANSB ? BN * BTS : BK * BS;
    constexpr int NT  = NCOL / BK;

    __shared__ float Xs[2][BM * XS];
    __shared__ float Bsh[2][BSZ];

    const int tid    = threadIdx.x;
    const int lane   = tid & 31;
    const int laneLo = lane & 15;
    const int laneHi = lane >> 4;
    const int wave   = tid >> 5;           // 8 waves
    const int wRow   = (wave >> 2) * 64;   // 2 waves in M
    const int wCol   = (wave & 3) * 32;    // 4 waves in N

    const int rowBase = blockIdx.y * BM;
    const int colBase = blockIdx.x * BN;

    v8f acc[4][2];
#pragma unroll
    for (int mi = 0; mi < 4; ++mi)
#pragma unroll
        for (int ni = 0; ni < 2; ++ni)
#pragma unroll
            for (int e = 0; e < 8; ++e) acc[mi][ni][e] = 0.0f;

#if HAVE_ASYNC_LDS
    // ---- CDNA5 async DMA staging: no VGPR staging, overlaps with WMMA ----
    auto prefetch = [&](int kt, int buf) {
        const int k0 = kt * BK;
#pragma unroll
        for (int s = 0; s < 2; ++s) {
            const int id = tid + s * 256;
            const int r = id >> 2, q = (id & 3) * 4;
            async_cp16(&X[(size_t)(rowBase + r) * NCOL + k0 + q],
                       &Xs[buf][r * XS + q]);
        }
        if (!TRANSB) {
#pragma unroll
            for (int s = 0; s < 2; ++s) {
                const int id = tid + s * 256;
                const int r = id >> 5, c = (id & 31) * 4;
                async_cp16(&W[(size_t)(k0 + r) * NCOL + colBase + c],
                           &Bsh[buf][r * BS + c]);
            }
        } else {
            // n-major tile: Bt[n][k] = W[(colBase+n)*512 + k0+k]; contiguous both sides
#pragma unroll
            for (int s = 0; s < 2; ++s) {
                const int id = tid + s * 256;
                const int c = id >> 2, rq = (id & 3) * 4;
                async_cp16(&W[(size_t)(colBase + c) * NCOL + k0 + rq],
                           &Bsh[buf][c * BTS + rq]);
            }
        }
    };
#else
    float4 px[2], pb[2];
    auto fetch = [&](int kt) {
        const int k0 = kt * BK;
#pragma unroll
        for (int s = 0; s < 2; ++s) {
            const int id = tid + s * 256;
            const int r = id >> 2, q = (id & 3) * 4;
            px[s] = *reinterpret_cast<const float4*>(
                &X[(size_t)(rowBase + r) * NCOL + k0 + q]);
        }
#pragma unroll
        for (int s = 0; s < 2; ++s) {
            const int id = tid + s * 256;
            if (!TRANSB) {
                const int r = id >> 5, c = (id & 31) * 4;
                pb[s] = *reinterpret_cast<const float4*>(
                    &W[(size_t)(k0 + r) * NCOL + colBase + c]);
            } else {
                const int c = id >> 2, rq = (id & 3) * 4;
                pb[s] = *reinterpret_cast<const float4*>(
                    &W[(size_t)(colBase + c) * NCOL + k0 + rq]);
            }
        }
    };
    auto stage = [&](int buf) {
#pragma unroll
        for (int s = 0; s < 2; ++s) {
            const int id = tid + s * 256;
            const int r = id >> 2, q = (id & 3) * 4;
            *reinterpret_cast<float4*>(&Xs[buf][r * XS + q]) = px[s];
        }
#pragma unroll
        for (int s = 0; s < 2; ++s) {
            const int id = tid + s * 256;
            if (!TRANSB) {
                const int r = id >> 5, c = (id & 31) * 4;
                *reinterpret_cast<float4*>(&Bsh[buf][r * BS + c]) = pb[s];
            } else {
                const int c = id >> 2, rq = (id & 3) * 4;
                *reinterpret_cast<float4*>(&Bsh[buf][c * BTS + rq]) = pb[s];
            }
        }
    };
#endif

    auto compute = [&](int buf) {
#pragma unroll
        for (int k4 = 0; k4 < 4; ++k4) {
            const int kb = k4 * 4;
            // A frag (16x4 f32): lane holds (m = lane%16, k = 2*(lane/16)+v), v=0,1
            v2f a[4];
#pragma unroll
            for (int mi = 0; mi < 4; ++mi)
                a[mi] = *reinterpret_cast<const v2f*>(
                    &Xs[buf][(wRow + mi * 16 + laneLo) * XS + kb + 2 * laneHi]);
            // B frag (4x16 f32): lane holds (k = v + 2*(lane/16), n = lane%16)
            v2f b[2];
#pragma unroll
            for (int ni = 0; ni < 2; ++ni) {
                if (TRANSB) {
                    // n-major tile: consecutive k pair -> single 8B LDS load
                    b[ni] = *reinterpret_cast<const v2f*>(
                        &Bsh[buf][(wCol + ni * 16 + laneLo) * BTS + kb + 2 * laneHi]);
                } else {
                    const float* p =
                        &Bsh[buf][(kb + 2 * laneHi) * BS + wCol + ni * 16 + laneLo];
                    b[ni][0] = p[0];
                    b[ni][1] = p[BS];
                }
            }
#pragma unroll
            for (int mi = 0; mi < 4; ++mi)
#pragma unroll
                for (int ni = 0; ni < 2; ++ni)
                    acc[mi][ni] = __builtin_amdgcn_wmma_f32_16x16x4_f32(
                        false, a[mi], false, b[ni], (short)0, acc[mi][ni],
                        false, false);
        }
    };

#if HAVE_ASYNC_LDS
    prefetch(0, 0);
    wait_async0();
    __syncthreads();
    for (int kt = 0; kt < NT; ++kt) {
        const int cur = kt & 1;
        if (kt + 1 < NT) prefetch(kt + 1, cur ^ 1);  // DMA next tile during compute
        compute(cur);
        if (kt + 1 < NT) wait_async0();
        __syncthreads();
    }
#else
    fetch(0);
    stage(0);
    __syncthreads();
    for (int kt = 0; kt < NT; ++kt) {
        const int cur = kt & 1;
        if (kt + 1 < NT) fetch(kt + 1);
        compute(cur);
        if (kt + 1 < NT) stage(cur ^ 1);
        __syncthreads();
    }
#endif

    // Epilogue: C/D layout -> lane holds (m = r + 8*(lane/16), n = lane%16) in VGPR r
#pragma unroll
    for (int mi = 0; mi < 4; ++mi)
#pragma unroll
        for (int ni = 0; ni < 2; ++ni) {
            const int gc = colBase + wCol + ni * 16 + laneLo;
#pragma unroll
            for (int r = 0; r < 8; ++r) {
                const int gr = rowBase + wRow + mi * 16 + laneHi * 8 + r;
                const size_t off = (size_t)gr * NCOL + gc;
                const float v = acc[mi][ni][r];
                C[off] = (EPI == 0) ? v : (D[off] - v);
            }
        }
}

// Monotonic RQ spline, in place: z <- z - y(z); logj[n] = sum_k logd.
// One wave per row n -> coalesced column reads, shuffle reduction, no atomics.
__global__ __launch_bounds__(256)
void rq_spline(const float* __restrict__ xx,
               const float* __restrict__ yy,
               const float* __restrict__ dl,
               float* __restrict__ z,
               float* __restrict__ logj)
{
    const int lane = threadIdx.x & 31;
    const int wave = threadIdx.x >> 5;
    const int n = blockIdx.x * 8 + wave;

    float lsum = 0.0f;
#pragma unroll 1
    for (int it = 0; it < NCOL / 32; ++it) {
        const int k = it * 32 + lane;
        const size_t e = (size_t)n * NCOL + k;
        const float x = z[e];
        const float* xr = &xx[k * MK];
        const float* yr = &yy[k * MK];
        const float* dr = &dl[k * MK];

        // searchsorted(xx[k], x, side='left'): first i with xr[i] >= x
        int lo = 0, hi = MK;
#pragma unroll
        for (int i = 0; i < 8; ++i) {   // 200 -> 1 in 8 halvings
            const int mid = (lo + hi) >> 1;
            const bool c = xr[mid] < x;
            lo = c ? mid + 1 : lo;
            hi = c ? hi : mid;
        }
        const int idx = lo;
        int b = idx < 1 ? 1 : (idx > MK - 1 ? MK - 1 : idx);
        b -= 1;

        const float xk = xr[b], xk1 = xr[b + 1];
        const float yk = yr[b], yk1 = yr[b + 1];
        const float dk = dr[b], dk1 = dr[b + 1];
        const float dx = xk1 - xk, dy = yk1 - yk;
        const float s = dy / dx;
        float xi = (x - xk) / dx;
        xi = fminf(fmaxf(xi, 0.0f), 1.0f);
        const float xi1 = 1.0f - xi;
        const float denom = s + (dk1 + dk - 2.0f * s) * xi * xi1;
        float y = yk + dy * (s * xi * xi + dk * xi * xi1) / denom;
        float logd = 2.0f * __logf(s)
                   + __logf(dk1 * xi * xi + 2.0f * s * xi * xi1 + dk * xi1 * xi1)
                   - 2.0f * __logf(denom);
        if (idx == 0) {                 // x <= xx[0] : linear tail
            y = yr[0] + dr[0] * (x - xr[0]);
            logd = __logf(dr[0]);
        } else if (idx == MK) {         // x > xx[M-1] : linear tail
            y = yr[MK - 1] + dr[MK - 1] * (x - xr[MK - 1]);
            logd = __logf(dr[MK - 1]);
        }
        lsum += logd;
        z[e] = x - y;                   // z = data0 - y
    }

#pragma unroll
    for (int off = 16; off > 0; off >>= 1)
        lsum += __shfl_down(lsum, off, 32);
    if (lane == 0) logj[n] = lsum;
}

} // namespace

extern "C" void kernel_launch(void* const* d_in, const int* in_sizes, int n_in,
                              void* d_out, int out_size, void* d_ws, size_t ws_size,
                              hipStream_t stream)
{
    const float* data  = (const float*)d_in[0];   // (N, 512)
    const float* A     = (const float*)d_in[1];   // (512, 512)
    const float* xx    = (const float*)d_in[2];   // (512, 200)
    const float* yy    = (const float*)d_in[3];
    const float* delta = (const float*)d_in[4];

    const int N = in_sizes[0] / NCOL;             // 65536

    float* out  = (float*)d_out;                  // (N, 512)
    float* logj = out + (size_t)N * NCOL;         // (N,)
    float* z    = (float*)d_ws;                   // N*512 f32 scratch

    const size_t zBytes  = (size_t)N * NCOL * sizeof(float);
    const size_t atBytes = (size_t)NCOL * NCOL * sizeof(float);
    const bool   haveAt  = ws_size >= zBytes + atBytes;   // deterministic per-launch

    dim3 grid(NCOL / 128, N / 128);               // (4, 512)

    if (haveAt) {
        // At = A^T (1 MB) -> both GEMMs use the n-major async path (b64 frags, no packing movs)
        float* At = (float*)((char*)d_ws + zBytes);
        transpose512<<<dim3(16, 16), dim3(32, 8), 0, stream>>>(A, At);
        // z = data @ A      (B[k][n] = A[k][n] = At[n*512+k])
        gemm512_wmma<true, 0><<<grid, 256, 0, stream>>>(data, At, nullptr, z);
    } else {
        gemm512_wmma<false, 0><<<grid, 256, 0, stream>>>(data, A, nullptr, z);
    }

    // z = z - y(z); logj = row-sum(logd)
    rq_spline<<<N / 8, 256, 0, stream>>>(xx, yy, delta, z, logj);

    // out = data - z @ A^T  (== remaining + y @ A^T);  B[k][n] = A[n][k] = A[n*512+k]
    gemm512_wmma<true, 1><<<grid, 256, 0, stream>>>(z, A, data, out);
}